// STEncoder_35622458753706
// MI455X (gfx1250) — compile-verified
//
#include <hip/hip_runtime.h>
#include <math.h>

#define Bn   32
#define Tin0 19
#define Nn   500
#define KTc  3
#define KSc  3
#define C0c  2
#define C1c  32
#define C2c  64
#define NQc  17
#define OUTL 11
#define EPSc 1e-5f

typedef __attribute__((ext_vector_type(2))) float v2f;
typedef __attribute__((ext_vector_type(8))) float v8f;

#if defined(__has_builtin)
#  if __has_builtin(__builtin_amdgcn_wmma_f32_16x16x4_f32)
#    define USE_WMMA_F32 1
#  endif
#endif
#ifndef USE_WMMA_F32
#  define USE_WMMA_F32 0
#endif

typedef __attribute__((address_space(3))) float lds_f;

// 32-bit LDS byte offset of a __shared__ object (generic -> AS3 -> int)
__device__ __forceinline__ unsigned lds_addr32(const void* p) {
    return (unsigned)(unsigned long long)(lds_f*)p;
}

// gfx1250 async DMA: global -> LDS, 16 bytes per active lane (ASYNCcnt-tracked)
__device__ __forceinline__ void async_copy_b128(unsigned ldsoff, const float* gptr) {
    asm volatile("global_load_async_to_lds_b128 %0, %1, off"
                 :: "v"(ldsoff), "v"(gptr)
                 : "memory");
}
__device__ __forceinline__ void wait_async0() {
    asm volatile("s_wait_asynccnt 0x0" ::: "memory");
}

__device__ __forceinline__ float sigmf(float x) { return 1.0f / (1.0f + __expf(-x)); }

// ---------------------------------------------------------------------------
// Stage 1: pad(T) + temporal conv (KT=3) + GLU.  x0:(B,19,N,2) -> Y:(B,32,19,N)
// ---------------------------------------------------------------------------
__global__ void glu_first(const float* __restrict__ x0, const float* __restrict__ W,
                          const float* __restrict__ bias, float* __restrict__ Yo) {
    int idx = blockIdx.x * blockDim.x + threadIdx.x;
    const int total = Bn * C1c * Tin0 * Nn;
    if (idx >= total) return;
    int n = idx % Nn;
    int t = (idx / Nn) % Tin0;
    int c = (idx / (Nn * Tin0)) % C1c;
    int b = idx / (Nn * Tin0 * C1c);
    float xc = bias[c], xg = bias[c + C1c];
    #pragma unroll
    for (int ci = 0; ci < C0c; ++ci) {
        #pragma unroll
        for (int dt = 0; dt < KTc; ++dt) {
            int ts = t + dt - (KTc - 1);       // padded time -> source time
            if (ts >= 0) {
                float xv = x0[(((size_t)b * Tin0 + ts) * Nn + n) * C0c + ci];
                xc += W[(c * C0c + ci) * KTc + dt] * xv;
                xg += W[((c + C1c) * C0c + ci) * KTc + dt] * xv;
            }
        }
    }
    float xin = (c < C0c) ? x0[(((size_t)b * Tin0 + t) * Nn + n) * C0c + c] : 0.0f;
    Yo[idx] = (xc + xin) * sigmf(xg);
}

// ---------------------------------------------------------------------------
// Pooler part A: S[b,q,l,n] = softmax_l( w_att@X + b_att )
// ---------------------------------------------------------------------------
__global__ void pool_att(const float* __restrict__ X, const float* __restrict__ Watt,
                         const float* __restrict__ batt, float* __restrict__ S) {
    int idx = blockIdx.x * blockDim.x + threadIdx.x;
    const int total = Bn * NQc * Nn;
    if (idx >= total) return;
    int n = idx % Nn;
    int q = (idx / Nn) % NQc;
    int b = idx / (Nn * NQc);
    float a[Tin0];
    float mx = -3.0e38f;
    for (int l = 0; l < Tin0; ++l) {
        float s = batt[q];
        #pragma unroll
        for (int c = 0; c < C1c; ++c)
            s += Watt[q * C1c + c] * X[(((size_t)b * C1c + c) * Tin0 + l) * Nn + n];
        a[l] = s;
        mx = fmaxf(mx, s);
    }
    float den = 0.0f;
    for (int l = 0; l < Tin0; ++l) { a[l] = __expf(a[l] - mx); den += a[l]; }
    float r = 1.0f / den;
    for (int l = 0; l < Tin0; ++l)
        S[(((size_t)b * NQc + q) * Tin0 + l) * Nn + n] = a[l] * r;
}

// Pooler part B: out[b,c,q,n] = relu( sum_l X[b,c,l,n]*S[b,q,l,n] + X[b,c,q+2,n] )
__global__ void pool_apply(const float* __restrict__ X, const float* __restrict__ S,
                           float* __restrict__ Yo) {
    int idx = blockIdx.x * blockDim.x + threadIdx.x;
    const int total = Bn * C1c * NQc * Nn;
    if (idx >= total) return;
    int n = idx % Nn;
    int q = (idx / Nn) % NQc;
    int c = (idx / (Nn * NQc)) % C1c;
    int b = idx / (Nn * NQc * C1c);
    float s = 0.0f;
    #pragma unroll
    for (int l = 0; l < Tin0; ++l)
        s += X[(((size_t)b * C1c + c) * Tin0 + l) * Nn + n] *
             S[(((size_t)b * NQc + q) * Tin0 + l) * Nn + n];
    s += X[(((size_t)b * C1c + c) * Tin0 + (q + KTc - 1)) * Nn + n];
    Yo[idx] = fmaxf(s, 0.0f);
}

// ---------------------------------------------------------------------------
// sconv helpers: U[b,o,t,m] = sum_c theta[c,o,k] * X[b,c,t,m]
// ---------------------------------------------------------------------------
__global__ void u_compute(const float* __restrict__ X, const float* __restrict__ theta,
                          float* __restrict__ U, int T, int kidx) {
    int idx = blockIdx.x * blockDim.x + threadIdx.x;
    const int total = Bn * C1c * T * Nn;
    if (idx >= total) return;
    int m = idx % Nn;
    int t = (idx / Nn) % T;
    int o = (idx / (Nn * T)) % C1c;
    int b = idx / (Nn * T * C1c);
    float s = 0.0f;
    #pragma unroll
    for (int c = 0; c < C1c; ++c)
        s += theta[(c * C1c + o) * KSc + kidx] *
             X[(((size_t)b * C1c + c) * T + t) * Nn + m];
    U[idx] = s;
}

__global__ void sconv_init(const float* __restrict__ X, const float* __restrict__ bsc,
                           float* __restrict__ Y, int T) {
    int idx = blockIdx.x * blockDim.x + threadIdx.x;
    const int total = Bn * C1c * T * Nn;
    if (idx >= total) return;
    int o = (idx / (Nn * T)) % C1c;
    Y[idx] = X[idx] + bsc[o];
}

__global__ void relu_ip(float* __restrict__ Y, int total) {
    int idx = blockIdx.x * blockDim.x + threadIdx.x;
    if (idx < total) Y[idx] = fmaxf(Y[idx], 0.0f);
}

// ---------------------------------------------------------------------------
// WMMA GEMM:  Y[b, row, n] += sum_m U[b, row, m] * Lk[b, kidx, n, m]
// rows = 32*T (multiple of 16).  Block = 128 thr (4 waves), tile 16(M) x 128(N).
// Each wave: 16x32 output via 2 independent accumulators (breaks the D->C
// dependency chain and reuses each A fragment twice).
// K staged in 64-wide LDS chunks, double-buffered via async global->LDS DMA
// (ASYNCcnt); out-of-range float4s zero-filled with plain LDS stores.
// ---------------------------------------------------------------------------
__global__ __launch_bounds__(128)
void sconv_gemm_wmma(const float* __restrict__ U, const float* __restrict__ Lk,
                     float* __restrict__ Y, int T, int kidx) {
    const int rows = C1c * T;
    const int m0 = blockIdx.x * 16;
    const int n0 = blockIdx.y * 128;
    const int b  = blockIdx.z;
    const int tid  = threadIdx.x;
    const int wave = tid >> 5;
    const int lane = tid & 31;
    const int laneCol = lane & 15;
    const int laneRow = (lane < 16) ? 0 : 8;
    const int t0 = (lane >> 4) << 1;           // K sub-offset: 0 or 2

    __shared__ __align__(16) float As[2][16][68];    // [buf][m][k]
    __shared__ __align__(16) float Bs[2][128][68];   // [buf][n][k]

    const float* Ubase = U + ((size_t)b * rows + m0) * Nn;
    const float* Lbase = Lk + (((size_t)b * KSc + kidx) * Nn) * Nn;

    // accumulate onto existing Y tile
    v8f acc0, acc1;
    #pragma unroll
    for (int v = 0; v < 8; ++v) {
        int r  = m0 + v + laneRow;
        int c0 = n0 + wave * 32 + laneCol;
        int c1 = c0 + 16;
        acc0[v] = (c0 < Nn) ? Y[((size_t)b * rows + r) * Nn + c0] : 0.0f;
        acc1[v] = (c1 < Nn) ? Y[((size_t)b * rows + r) * Nn + c1] : 0.0f;
    }

    auto stage = [&](int buf, int k0) {
        // A tile 16x64: 2 float4 per thread, coalesced along K
        #pragma unroll
        for (int i = 0; i < 2; ++i) {
            int flat = i * 512 + tid * 4;
            int r = flat >> 6, kk = flat & 63;
            int gK = k0 + kk;
            if (gK < Nn) {
                async_copy_b128(lds_addr32(&As[buf][r][kk]), Ubase + (size_t)r * Nn + gK);
            } else {
                *(float4*)&As[buf][r][kk] = make_float4(0.f, 0.f, 0.f, 0.f);
            }
        }
        // B tile 128(n) x 64(k): Bs[n][k] = Lk[b,kidx, n0+n, k0+k] (contig in k)
        #pragma unroll
        for (int i = 0; i < 16; ++i) {
            int flat = i * 512 + tid * 4;
            int nl = flat >> 6, kk = flat & 63;
            int gN = n0 + nl, gK = k0 + kk;
            if (gN < Nn && gK < Nn) {
                async_copy_b128(lds_addr32(&Bs[buf][nl][kk]),
                                Lbase + (size_t)gN * Nn + gK);
            } else {
                *(float4*)&Bs[buf][nl][kk] = make_float4(0.f, 0.f, 0.f, 0.f);
            }
        }
    };

    stage(0, 0);
    wait_async0();
    __syncthreads();

    for (int chunk = 0; chunk < 8; ++chunk) {
        int cur = chunk & 1;
        if (chunk + 1 < 8) stage(cur ^ 1, (chunk + 1) * 64);   // overlap DMA

        #pragma unroll
        for (int kk0 = 0; kk0 < 64; kk0 += 4) {
            v2f a, b0, b1;
            a.x  = As[cur][laneCol][kk0 + t0];
            a.y  = As[cur][laneCol][kk0 + t0 + 1];
            b0.x = Bs[cur][wave * 32 + laneCol][kk0 + t0];
            b0.y = Bs[cur][wave * 32 + laneCol][kk0 + t0 + 1];
            b1.x = Bs[cur][wave * 32 + 16 + laneCol][kk0 + t0];
            b1.y = Bs[cur][wave * 32 + 16 + laneCol][kk0 + t0 + 1];
#if USE_WMMA_F32
            acc0 = __builtin_amdgcn_wmma_f32_16x16x4_f32(
                false, a, false, b0, (short)0, acc0, false, false);
            acc1 = __builtin_amdgcn_wmma_f32_16x16x4_f32(
                false, a, false, b1, (short)0, acc1, false, false);
#else
            acc0[0] = fmaf(a.x, b0.x, acc0[0]);
            acc1[0] = fmaf(a.y, b1.x, acc1[0]);
#endif
        }
        wait_async0();       // next buffer fully resident before barrier
        __syncthreads();     // all waves done reading cur / writing nxt
    }

    #pragma unroll
    for (int v = 0; v < 8; ++v) {
        int r  = m0 + v + laneRow;
        int c0 = n0 + wave * 32 + laneCol;
        int c1 = c0 + 16;
        if (c0 < Nn) Y[((size_t)b * rows + r) * Nn + c0] = acc0[v];
        if (c1 < Nn) Y[((size_t)b * rows + r) * Nn + c1] = acc1[v];
    }
}

// ---------------------------------------------------------------------------
// LayerNorm over (N, C) per (b, t); in-place on (B, C, T, N); gamma/beta: (N, C)
// ---------------------------------------------------------------------------
__global__ __launch_bounds__(256)
void ln_nc_inplace(float* __restrict__ X, const float* __restrict__ gamma,
                   const float* __restrict__ beta, int C, int T) {
    int b = blockIdx.x / T, t = blockIdx.x % T;
    const int cnt = C * Nn;
    float s = 0.0f, s2 = 0.0f;
    for (int i = threadIdx.x; i < cnt; i += 256) {
        int c = i / Nn, n = i % Nn;
        float v = X[(((size_t)b * C + c) * T + t) * Nn + n];
        s += v; s2 += v * v;
    }
    __shared__ float r0[256], r1[256];
    r0[threadIdx.x] = s; r1[threadIdx.x] = s2;
    __syncthreads();
    for (int off = 128; off > 0; off >>= 1) {
        if (threadIdx.x < off) {
            r0[threadIdx.x] += r0[threadIdx.x + off];
            r1[threadIdx.x] += r1[threadIdx.x + off];
        }
        __syncthreads();
    }
    float mu  = r0[0] / (float)cnt;
    float var = r1[0] / (float)cnt - mu * mu;
    float rstd = rsqrtf(var + EPSc);
    for (int i = threadIdx.x; i < cnt; i += 256) {
        int c = i / Nn, n = i % Nn;
        size_t p = (((size_t)b * C + c) * T + t) * Nn + n;
        X[p] = (X[p] - mu) * rstd * gamma[n * C + c] + beta[n * C + c];
    }
}

// ---------------------------------------------------------------------------
// relu temporal layer: out[b,co,t,n] = relu(conv3 + residual), Cin=32 -> Cout=64
// ---------------------------------------------------------------------------
__global__ void tconv_relu(const float* __restrict__ X, const float* __restrict__ W,
                           const float* __restrict__ bias, float* __restrict__ Yo,
                           int Tin) {
    const int Tout = Tin - (KTc - 1);
    int idx = blockIdx.x * blockDim.x + threadIdx.x;
    const int total = Bn * C2c * Tout * Nn;
    if (idx >= total) return;
    int n  = idx % Nn;
    int t  = (idx / Nn) % Tout;
    int co = (idx / (Nn * Tout)) % C2c;
    int b  = idx / (Nn * Tout * C2c);
    float s = bias[co];
    for (int ci = 0; ci < C1c; ++ci) {
        #pragma unroll
        for (int dt = 0; dt < KTc; ++dt)
            s += W[(co * C1c + ci) * KTc + dt] *
                 X[(((size_t)b * C1c + ci) * Tin + t + dt) * Nn + n];
    }
    if (co < C1c) s += X[(((size_t)b * C1c + co) * Tin + t + KTc - 1) * Nn + n];
    Yo[idx] = fmaxf(s, 0.0f);
}

// ---------------------------------------------------------------------------
// GLU mid layer: Cin=64 -> Cout=32, with 1x1 align projection residual
// ---------------------------------------------------------------------------
__global__ void glu_mid(const float* __restrict__ X, const float* __restrict__ W,
                        const float* __restrict__ bias, const float* __restrict__ Wal,
                        const float* __restrict__ bal, float* __restrict__ Yo, int Tin) {
    const int Tout = Tin - (KTc - 1);
    int idx = blockIdx.x * blockDim.x + threadIdx.x;
    const int total = Bn * C1c * Tout * Nn;
    if (idx >= total) return;
    int n = idx % Nn;
    int t = (idx / Nn) % Tout;
    int c = (idx / (Nn * Tout)) % C1c;
    int b = idx / (Nn * Tout * C1c);
    float xc = bias[c], xg = bias[c + C1c];
    float xin = bal[c];
    for (int ci = 0; ci < C2c; ++ci) {
        float v0 = X[(((size_t)b * C2c + ci) * Tin + t + 0) * Nn + n];
        float v1 = X[(((size_t)b * C2c + ci) * Tin + t + 1) * Nn + n];
        float v2 = X[(((size_t)b * C2c + ci) * Tin + t + 2) * Nn + n];
        const float* wc = &W[(c * C2c + ci) * KTc];
        const float* wg = &W[((c + C1c) * C2c + ci) * KTc];
        xc += wc[0] * v0 + wc[1] * v1 + wc[2] * v2;
        xg += wg[0] * v0 + wg[1] * v1 + wg[2] * v2;
        xin += Wal[c * C2c + ci] * v2;
    }
    Yo[idx] = (xc + xin) * sigmf(xg);
}

// ---------------------------------------------------------------------------
// Output GLU: full-width temporal conv (kt = OUT_LEN = 11), Cin=Cout=64
// X:(B,64,11,N) -> Yo:(B,64,N)
// ---------------------------------------------------------------------------
__global__ void glu_final(const float* __restrict__ X, const float* __restrict__ W,
                          const float* __restrict__ bias, float* __restrict__ Yo) {
    int idx = blockIdx.x * blockDim.x + threadIdx.x;
    const int total = Bn * C2c * Nn;
    if (idx >= total) return;
    int n = idx % Nn;
    int c = (idx / Nn) % C2c;
    int b = idx / (Nn * C2c);
    float xc = bias[c], xg = bias[c + C2c];
    for (int ci = 0; ci < C2c; ++ci) {
        #pragma unroll
        for (int dt = 0; dt < OUTL; ++dt) {
            float xv = X[(((size_t)b * C2c + ci) * OUTL + dt) * Nn + n];
            xc += W[(c * C2c + ci) * OUTL + dt] * xv;
            xg += W[((c + C2c) * C2c + ci) * OUTL + dt] * xv;
        }
    }
    float xin = X[(((size_t)b * C2c + c) * OUTL + (OUTL - 1)) * Nn + n];
    Yo[idx] = (xc + xin) * sigmf(xg);
}

// Final LayerNorm over (N, C2) per b, output transposed to (B, N, C2)
__global__ __launch_bounds__(256)
void ln_final(const float* __restrict__ X, const float* __restrict__ gamma,
              const float* __restrict__ beta, float* __restrict__ Out) {
    int b = blockIdx.x;
    const int cnt = C2c * Nn;
    float s = 0.0f, s2 = 0.0f;
    for (int i = threadIdx.x; i < cnt; i += 256) {
        float v = X[(size_t)b * cnt + i];
        s += v; s2 += v * v;
    }
    __shared__ float r0[256], r1[256];
    r0[threadIdx.x] = s; r1[threadIdx.x] = s2;
    __syncthreads();
    for (int off = 128; off > 0; off >>= 1) {
        if (threadIdx.x < off) {
            r0[threadIdx.x] += r0[threadIdx.x + off];
            r1[threadIdx.x] += r1[threadIdx.x + off];
        }
        __syncthreads();
    }
    float mu  = r0[0] / (float)cnt;
    float var = r1[0] / (float)cnt - mu * mu;
    float rstd = rsqrtf(var + EPSc);
    for (int i = threadIdx.x; i < cnt; i += 256) {
        int c = i / Nn, n = i % Nn;
        float v = X[(size_t)b * cnt + i];
        Out[((size_t)b * Nn + n) * C2c + c] =
            (v - mu) * rstd * gamma[n * C2c + c] + beta[n * C2c + c];
    }
}

// ---------------------------------------------------------------------------
static inline int cdiv_i(long long a, int b) { return (int)((a + b - 1) / b); }

static void run_sconv(const float* Xin, const float* Lk, const float* theta,
                      const float* bsc, float* Ubuf, float* Ybuf, int T,
                      hipStream_t stream) {
    const int total = Bn * C1c * T * Nn;
    sconv_init<<<cdiv_i(total, 256), 256, 0, stream>>>(Xin, bsc, Ybuf, T);
    for (int k = 0; k < KSc; ++k) {
        u_compute<<<cdiv_i(total, 256), 256, 0, stream>>>(Xin, theta, Ubuf, T, k);
        dim3 grid((C1c * T) / 16, 4, Bn);
        sconv_gemm_wmma<<<grid, 128, 0, stream>>>(Ubuf, Lk, Ybuf, T, k);
    }
    relu_ip<<<cdiv_i(total, 256), 256, 0, stream>>>(Ybuf, total);
}

extern "C" void kernel_launch(void* const* d_in, const int* in_sizes, int n_in,
                              void* d_out, int out_size, void* d_ws, size_t ws_size,
                              hipStream_t stream) {
    (void)in_sizes; (void)n_in; (void)out_size; (void)ws_size;
    const float* x0    = (const float*)d_in[0];
    const float* Lk    = (const float*)d_in[1];
    const float* w11   = (const float*)d_in[2];
    const float* b11   = (const float*)d_in[3];
    const float* watt  = (const float*)d_in[4];
    const float* batt  = (const float*)d_in[5];
    const float* th12  = (const float*)d_in[6];
    const float* bsc12 = (const float*)d_in[7];
    const float* gls1  = (const float*)d_in[8];
    const float* bls1  = (const float*)d_in[9];
    const float* w13   = (const float*)d_in[10];
    const float* b13   = (const float*)d_in[11];
    const float* gl1   = (const float*)d_in[12];
    const float* bl1   = (const float*)d_in[13];
    const float* w21   = (const float*)d_in[14];
    const float* b21   = (const float*)d_in[15];
    const float* wal   = (const float*)d_in[16];
    const float* bal   = (const float*)d_in[17];
    const float* th22  = (const float*)d_in[18];
    const float* bsc22 = (const float*)d_in[19];
    const float* gls2  = (const float*)d_in[20];
    const float* bls2  = (const float*)d_in[21];
    const float* w23   = (const float*)d_in[22];
    const float* b23   = (const float*)d_in[23];
    const float* gl2   = (const float*)d_in[24];
    const float* bl2   = (const float*)d_in[25];
    const float* wout  = (const float*)d_in[26];
    const float* bout  = (const float*)d_in[27];
    const float* gl3   = (const float*)d_in[28];
    const float* bl3   = (const float*)d_in[29];

    // workspace slots (max usage: A 61.44MB, B 45.06MB, C 34.82MB)
    char* ws = (char*)d_ws;
    float* bufA = (float*)(ws);
    float* bufB = (float*)(ws + 61440000);
    float* bufC = (float*)(ws + 61440000 + 45056000);

    // 1) GLU layer 1: x0 -> bufA (B,32,19,N)
    {
        int total = Bn * C1c * Tin0 * Nn;
        glu_first<<<cdiv_i(total, 256), 256, 0, stream>>>(x0, w11, b11, bufA);
    }
    // 2) Pooler: bufA -> bufB (B,32,17,N); softmax scratch in bufC
    {
        int ta = Bn * NQc * Nn;
        pool_att<<<cdiv_i(ta, 256), 256, 0, stream>>>(bufA, watt, batt, bufC);
        int tb = Bn * C1c * NQc * Nn;
        pool_apply<<<cdiv_i(tb, 256), 256, 0, stream>>>(bufA, bufC, bufB);
    }
    // 3) sconv1: bufB -> bufC (U scratch in bufA)
    run_sconv(bufB, Lk, th12, bsc12, bufA, bufC, NQc, stream);
    // 4) LN (N,C1) in-place on bufC
    ln_nc_inplace<<<Bn * NQc, 256, 0, stream>>>(bufC, gls1, bls1, C1c, NQc);
    // 5) relu temporal layer: bufC (32,17) -> bufA (64,15)
    {
        int total = Bn * C2c * (NQc - 2) * Nn;
        tconv_relu<<<cdiv_i(total, 256), 256, 0, stream>>>(bufC, w13, b13, bufA, NQc);
    }
    // 6) LN (N,C2) in-place on bufA
    ln_nc_inplace<<<Bn * (NQc - 2), 256, 0, stream>>>(bufA, gl1, bl1, C2c, NQc - 2);
    // 7) GLU layer 2: bufA (64,15) -> bufB (32,13)
    {
        int total = Bn * C1c * 13 * Nn;
        glu_mid<<<cdiv_i(total, 256), 256, 0, stream>>>(bufA, w21, b21, wal, bal, bufB, 15);
    }
    // 8) sconv2: bufB -> bufC (U scratch in bufA)
    run_sconv(bufB, Lk, th22, bsc22, bufA, bufC, 13, stream);
    // 9) LN (N,C1) in-place on bufC
    ln_nc_inplace<<<Bn * 13, 256, 0, stream>>>(bufC, gls2, bls2, C1c, 13);
    // 10) relu temporal layer: bufC (32,13) -> bufB (64,11)
    {
        int total = Bn * C2c * 11 * Nn;
        tconv_relu<<<cdiv_i(total, 256), 256, 0, stream>>>(bufC, w23, b23, bufB, 13);
    }
    // 11) LN (N,C2) in-place on bufB
    ln_nc_inplace<<<Bn * 11, 256, 0, stream>>>(bufB, gl2, bl2, C2c, 11);
    // 12) Output GLU (kt = 11): bufB -> bufC (B,64,N)
    {
        int total = Bn * C2c * Nn;
        glu_final<<<cdiv_i(total, 256), 256, 0, stream>>>(bufB, wout, bout, bufC);
    }
    // 13) Final LN -> d_out (B, 1, N, 64)
    ln_final<<<Bn, 256, 0, stream>>>(bufC, gl3, bl3, (float*)d_out);
}